// MMFRamanAmplifier_74182675137128
// MI455X (gfx1250) — compile-verified
//
#include <hip/hip_runtime.h>
#include <hip/hip_bf16.h>

typedef __attribute__((ext_vector_type(2))) float v2f;
typedef __attribute__((ext_vector_type(8))) float v8f;

#define NPUMP 4
#define NMODE 4
#define NCHAN 100
#define NF    (NPUMP + NCHAN)        // 104 frequencies
#define NFM   (NF * NMODE)           // 416 state entries
#define FPAD  112                    // F padded to multiple of 16
#define GSTR  116                    // LDS row stride (bank-conflict-free, even)
#define NKC   (FPAD / 4)             // 28 K-chunks of 4 for wmma f32 16x16x4
#define NTHREADS 224                 // 7 waves: one 16-row output tile each
#define C0F   299792458.0f
#define ALPHA_LIN 2.3025850929940457e-4f  // 1e-3*ln(10)/10

__global__ __launch_bounds__(NTHREADS)
void raman_rk4_kernel(const float* __restrict__ x,
                      const float* __restrict__ sfreq,
                      const float* __restrict__ spow,
                      const float* __restrict__ sloss,
                      const float* __restrict__ lossc,
                      const float* __restrict__ ovg,
                      const float* __restrict__ rr,
                      const int*   __restrict__ stepsPtr,
                      const float* __restrict__ lengthPtr,
                      const float* __restrict__ maxfPtr,
                      float* __restrict__ out,
                      int Lr)
{
  extern __shared__ float smem[];
  float* gainL  = smem;                       // FPAD*GSTR  (zero-padded)
  float* QtL    = gainL + FPAD*GSTR;          // 16*GSTR    (rows 4..15 stay zero)
  float* freqsL = QtL + 16*GSTR;              // FPAD
  float* alphaL = freqsL + FPAD;              // NFM
  float* PL     = alphaL + NFM;               // NFM  current state
  float* YcurL  = PL + NFM;                   // NFM  current eval point
  float* dPL    = YcurL + NFM;                // NFM  current k
  float* KaccL  = dPL + NFM;                  // NFM  k1+2k2+2k3

  const int b = blockIdx.x;
  const int t = threadIdx.x;
  const float* xb = x + b * (NPUMP * (1 + NMODE));

  // mode-overlap matrix in registers (uniform -> scalar loads)
  float ov[16];
  #pragma unroll
  for (int k = 0; k < 16; ++k) ov[k] = ovg[k];

  const int   nsteps = stepsPtr[0];
  const float length = lengthPtr[0];
  const float maxf   = maxfPtr[0];
  const float h      = length / (float)(nsteps - 1);

  // zero padded regions (gain K/M padding, Qt N padding)
  for (int idx = t; idx < FPAD*GSTR; idx += NTHREADS) gainL[idx] = 0.0f;
  for (int idx = t; idx < 16*GSTR;  idx += NTHREADS) QtL[idx]   = 0.0f;

  // frequencies, losses, initial powers
  if (t < NF) {
    float f;
    if (t < NPUMP) {
      float wl = xb[t];
      f = C0F / wl;
      float wlnm = wl * 1e9f;
      float pl = (lossc[2] + lossc[1]*wlnm + lossc[0]*wlnm*wlnm) * ALPHA_LIN;
      #pragma unroll
      for (int n = 0; n < NMODE; ++n) {
        alphaL[t*NMODE + n] = pl;
        PL[t*NMODE + n]     = xb[NPUMP + t*NMODE + n];
      }
    } else {
      int c = t - NPUMP;
      f = sfreq[c];
      #pragma unroll
      for (int n = 0; n < NMODE; ++n) {
        alphaL[t*NMODE + n] = sloss[c*NMODE + n];
        PL[t*NMODE + n]     = spow[c*NMODE + n];
      }
    }
    freqsL[t] = f;
  }
  __syncthreads();

  // gain[i,j]: interpolated antisymmetric Raman gain with photon-number scaling
  {
    const float scalepos = (float)(Lr - 1) / maxf;
    for (int idx = t; idx < NF*NF; idx += NTHREADS) {
      int i = idx / NF;
      int j = idx - i*NF;
      float fi = freqsL[i];
      float fj = freqsL[j];
      float fd = fj - fi;
      float pos = fabsf(fd) * scalepos;
      int i0 = (int)pos;
      i0 = i0 < (Lr - 2) ? i0 : (Lr - 2);
      float w  = pos - (float)i0;
      float g0 = rr[i0], g1 = rr[i0 + 1];   // L2-resident, setup only
      float g  = g0 + (g1 - g0) * w;
      g = (fd < 0.0f) ? -g : g;
      g *= fmaxf(1.0f, fi / fj);
      gainL[i*GSTR + j] = g;
    }
  }
  __syncthreads();

  // initial eval point Y=P and Qt[m][j] = overlap row m . P[j,:]
  if (t < NF) {
    int j = t;
    float y[NMODE];
    #pragma unroll
    for (int n = 0; n < NMODE; ++n) { y[n] = PL[j*NMODE + n]; YcurL[j*NMODE + n] = y[n]; }
    #pragma unroll
    for (int m = 0; m < NMODE; ++m)
      QtL[m*GSTR + j] = ov[m*4+0]*y[0] + ov[m*4+1]*y[1] + ov[m*4+2]*y[2] + ov[m*4+3]*y[3];
  }
  __syncthreads();

  // per-lane fragment addressing (ISA 16x4 f32 A layout; B mirrored)
  const int wv   = t >> 5;
  const int ln   = t & 31;
  const int koff = (ln >> 4) << 1;                 // K=0/1 (low lanes) or K=2/3
  const float* gaRow = gainL + ((wv << 4) + (ln & 15)) * GSTR + koff;  // A: row M=16w+lane
  const float* qbRow = QtL   + (ln & 15) * GSTR + koff;               // B: col N=lane
  const int mcol = ln & 15;
  const int ibase = (wv << 4) + ((ln >> 4) << 3);  // C/D: rows v(+8) per lane half

  for (int s = 0; s < nsteps - 1; ++s) {
    #pragma unroll
    for (int stage = 0; stage < 4; ++stage) {
      // ---- k = ode(Ycur): R = gain * Q via f32 WMMA, then elementwise ----
      v8f acc = {};
      for (int kc = 0; kc < NKC; ++kc) {
        v2f a  = *(const v2f*)(gaRow + kc*4);
        v2f bq = *(const v2f*)(qbRow + kc*4);
        acc = __builtin_amdgcn_wmma_f32_16x16x4_f32(
            false, a, false, bq, (short)0, acc, false, false);
      }
      if (mcol < NMODE) {
        #pragma unroll
        for (int v = 0; v < 8; ++v) {
          int i = ibase + v;
          if (i < NF) {
            int e = i*NMODE + mcol;
            dPL[e] = (acc[v] - alphaL[e]) * YcurL[e];   // (-alpha + G P) * P
          }
        }
      }
      __syncthreads();

      // ---- RK4 stage combine + next Qt (one freq index per thread) ----
      if (t < NF) {
        int j = t;
        float y[NMODE];
        if (stage < 3) {
          float cs = (stage == 2) ? h : 0.5f * h;
          #pragma unroll
          for (int n = 0; n < NMODE; ++n) {
            int e = j*NMODE + n;
            float kv = dPL[e];
            if (stage == 0) KaccL[e] = kv;
            else            KaccL[e] += 2.0f * kv;
            float yy = PL[e] + cs * kv;
            YcurL[e] = yy; y[n] = yy;
          }
        } else {
          float h6 = h * (1.0f / 6.0f);
          #pragma unroll
          for (int n = 0; n < NMODE; ++n) {
            int e = j*NMODE + n;
            float pn = PL[e] + h6 * (KaccL[e] + dPL[e]);
            PL[e] = pn; YcurL[e] = pn; y[n] = pn;
          }
        }
        #pragma unroll
        for (int m = 0; m < NMODE; ++m)
          QtL[m*GSTR + j] = ov[m*4+0]*y[0] + ov[m*4+1]*y[1] + ov[m*4+2]*y[2] + ov[m*4+3]*y[3];
      }
      __syncthreads();
    }
  }

  // signal spectrum only: drop the first NPUMP*NMODE entries
  for (int e = t; e < NFM; e += NTHREADS) {
    if (e >= NPUMP * NMODE)
      out[b * (NCHAN * NMODE) + (e - NPUMP * NMODE)] = PL[e];
  }
}

extern "C" void kernel_launch(void* const* d_in, const int* in_sizes, int n_in,
                              void* d_out, int out_size, void* d_ws, size_t ws_size,
                              hipStream_t stream) {
  const float* x     = (const float*)d_in[0];
  const float* sfreq = (const float*)d_in[1];
  const float* spow  = (const float*)d_in[2];
  const float* sloss = (const float*)d_in[3];
  const float* lossc = (const float*)d_in[4];
  const float* ovg   = (const float*)d_in[5];
  const float* rr    = (const float*)d_in[6];
  const int*   stepsP= (const int*)d_in[10];
  const float* lenP  = (const float*)d_in[11];
  const float* maxfP = (const float*)d_in[12];
  float* out = (float*)d_out;

  const int B  = in_sizes[0] / (NPUMP * (1 + NMODE));
  const int Lr = in_sizes[6];

  const size_t shmem =
      (size_t)(FPAD*GSTR + 16*GSTR + FPAD + 5*NFM) * sizeof(float);  // ~68 KB

  raman_rk4_kernel<<<B, NTHREADS, shmem, stream>>>(
      x, sfreq, spow, sloss, lossc, ovg, rr, stepsP, lenP, maxfP, out, Lr);
}